// Net_SLSTM_29910152249483
// MI455X (gfx1250) — compile-verified
//
#include <hip/hip_runtime.h>

typedef __attribute__((ext_vector_type(2))) float v2f;
typedef __attribute__((ext_vector_type(8))) float v8f;

constexpr int kT   = 1000;   // timesteps
constexpr int kB   = 256;    // batch
constexpr int kDin = 14;     // layer-1 input features
constexpr int kH   = 128;    // hidden
constexpr int kG   = 512;    // 4*H gate width
constexpr int kBT  = 16;     // batch tile per workgroup
constexpr int kNWG = kB / kBT;        // 16 recurrent workgroups
constexpr int kPS  = 1056;   // pair-row stride (floats): [kpair][2*512 + pad]; 1056%64==32 -> banks split
constexpr int kAS  = 132;    // padded A-tile row stride [16][128] (stride%64 == 4)
constexpr float kBnEps = 1e-5f;

__device__ __forceinline__ float sigmf(float v) { return 1.0f / (1.0f + __expf(-v)); }

// D = A(16x4 f32) * B(4x16 f32) + C  -- wave32, per CDNA5 ISA
__device__ __forceinline__ v8f wmma4(v2f a, v2f b, v8f c) {
    return __builtin_amdgcn_wmma_f32_16x16x4_f32(false, a, false, b, (short)0, c, false, false);
}

// Async global->LDS copies (ASYNCcnt path, CDNA5 ISA).
// LDS address = low 32 bits of the generic shared pointer (flat-LDS truncation, §10.2).
__device__ __forceinline__ unsigned lds_off(const void* p) {
    return (unsigned)(unsigned long long)p;
}
__device__ __forceinline__ void async_b32(unsigned ldsoff, const float* g) {
    asm volatile("global_load_async_to_lds_b32 %0, %1, off" :: "v"(ldsoff), "v"(g) : "memory");
}
__device__ __forceinline__ void async_b128(unsigned ldsoff, const float* g) {
    asm volatile("global_load_async_to_lds_b128 %0, %1, off" :: "v"(ldsoff), "v"(g) : "memory");
}
__device__ __forceinline__ void async_wait0() {
    asm volatile("s_wait_asynccnt 0" ::: "memory");
}

// -------------------------------------------------------------------------
// Layer-1 SLSTM recurrence. One WG per 16 batch rows; wave w owns h-block w
// and all 4 gate tiles for it. w_hh^T lives in LDS (k-pair interleaved so a
// B-fragment is one ds_load_b64); syn/mem state stays in VGPRs.
// -------------------------------------------------------------------------
__global__ __launch_bounds__(256) void slstm_layer1(
    const float* __restrict__ x,     // [T,B,14]
    const float* __restrict__ w_ih,  // [512,14]
    const float* __restrict__ w_hh,  // [512,128]
    const float* __restrict__ b_ih,  // [512]
    const float* __restrict__ b_hh,  // [512]
    const float* __restrict__ thr_p, // [1]
    float* __restrict__ spk1,        // [T*B,128] out
    float* __restrict__ bn_sum)      // [128] spike counts (atomic)
{
    __shared__ float lds_whh[(kH / 2) * kPS];  // w_hh^T pair-interleaved, 264 KB
    __shared__ float lds_wih[8 * kPS];         // w_ih^T padded K->16, 33 KB
    __shared__ float lds_mem[kBT * kAS];       // recurrent state as A-matrix
    __shared__ float lds_x[kBT * 20];          // x_t tile padded K->16

    const int tid  = threadIdx.x;
    const int lane = tid & 31;
    const int wv   = tid >> 5;      // 0..7 == h-block
    const int half = lane >> 4;
    const int l16  = lane & 15;
    const int b0   = blockIdx.x * kBT;
    const float thr = thr_p[0];

    for (int i = tid; i < kG * kH; i += 256) {     // w_hh^T, pair-interleaved
        int n = i / kH, k = i % kH;
        lds_whh[(k >> 1) * kPS + 2 * n + (k & 1)] = w_hh[i];
    }
    for (int i = tid; i < 16 * kG; i += 256) {     // w_ih^T, pad K to 16
        int k = i / kG, n = i % kG;
        lds_wih[(k >> 1) * kPS + 2 * n + (k & 1)] = (k < kDin) ? w_ih[n * kDin + k] : 0.0f;
    }
    for (int i = tid; i < kBT * kAS; i += 256) lds_mem[i] = 0.0f;
    for (int i = tid; i < kBT * 20;  i += 256) lds_x[i] = 0.0f;  // pad cols stay 0

    float bias[4];
    for (int q = 0; q < 4; ++q) {
        int col = q * kH + wv * 16 + l16;
        bias[q] = b_ih[col] + b_hh[col];
    }
    float syn[8], memv[8], cnt = 0.0f;
    for (int r = 0; r < 8; ++r) { syn[r] = 0.0f; memv[r] = 0.0f; }
    __syncthreads();

    for (int t = 0; t < kT; ++t) {
        // async-stage x_t tile (16 rows x 14 floats, coalesced)
        if (tid < kBT * kDin) {
            int r = tid / kDin, k = tid % kDin;
            async_b32(lds_off(&lds_x[r * 20 + k]),
                      &x[(size_t)(t * kB + b0 + r) * kDin + k]);
        }
        async_wait0();
        __syncthreads();

        v8f acc[4];
        for (int q = 0; q < 4; ++q)
            for (int r = 0; r < 8; ++r) acc[q][r] = bias[q];

        // input projection: K=16 (4 k-slices)
        #pragma unroll
        for (int ks = 0; ks < 4; ++ks) {
            v2f a = *(const v2f*)&lds_x[l16 * 20 + ks * 4 + 2 * half];
            #pragma unroll
            for (int q = 0; q < 4; ++q) {
                int n = q * kH + wv * 16 + l16;
                v2f b = *(const v2f*)&lds_wih[(ks * 2 + half) * kPS + 2 * n];
                acc[q] = wmma4(a, b, acc[q]);
            }
        }
        // recurrent projection: K=128 (32 k-slices)
        #pragma unroll 4
        for (int ks = 0; ks < 32; ++ks) {
            v2f a = *(const v2f*)&lds_mem[l16 * kAS + ks * 4 + 2 * half];
            #pragma unroll
            for (int q = 0; q < 4; ++q) {
                int n = q * kH + wv * 16 + l16;
                v2f b = *(const v2f*)&lds_whh[(ks * 2 + half) * kPS + 2 * n];
                acc[q] = wmma4(a, b, acc[q]);
            }
        }
        __syncthreads();  // all lds_mem / lds_x reads complete

        #pragma unroll
        for (int r = 0; r < 8; ++r) {
            float ig = sigmf(acc[0][r]);
            float fg = sigmf(acc[1][r]);
            float gg = tanhf(acc[2][r]);
            float og = sigmf(acc[3][r]);
            float s  = fg * syn[r] + ig * gg;
            float reset = (memv[r] - thr > 0.0f) ? thr : 0.0f;  // from PREVIOUS mem
            float mnew  = og * tanhf(s) - reset;
            float spike = (mnew - thr > 0.0f) ? 1.0f : 0.0f;
            syn[r] = s; memv[r] = mnew;
            int row = r + 8 * half;
            lds_mem[row * kAS + wv * 16 + l16] = mnew;
            spk1[(size_t)(t * kB + b0 + row) * kH + wv * 16 + l16] = spike;
            cnt += spike;
        }
        __syncthreads();  // lds_mem writes visible before next step
    }
    atomicAdd(&bn_sum[wv * 16 + l16], cnt);
}

// -------------------------------------------------------------------------
// BN finalize + layer-2 projection bias. Spikes are {0,1} => var = m - m^2.
// Fold BN: (s*a + bb) @ W^T = s @ (diag(a) W^T) + bb @ W^T.
//   bn_a[k]  -> scales W^T rows at LDS staging time in proj2_gemm
//   bbrow[n] -> b_ih2[n]+b_hh2[n] + sum_k bb[k]*w_ih2[n,k]  (accumulator init)
// -------------------------------------------------------------------------
__global__ __launch_bounds__(512) void bn_proj_bias(
    const float* __restrict__ bn_sum, const float* __restrict__ gamma,
    const float* __restrict__ beta,
    const float* __restrict__ b_ih2, const float* __restrict__ b_hh2,
    const float* __restrict__ w_ih2,
    float* __restrict__ bn_a, float* __restrict__ bbrow)
{
    __shared__ float s_bb[kH];
    int tid = threadIdx.x;
    if (tid < kH) {
        float mean = bn_sum[tid] * (1.0f / (float)(kT * kB));
        float var  = mean - mean * mean;
        float a = gamma[tid] * rsqrtf(var + kBnEps);
        bn_a[tid] = a;
        s_bb[tid] = beta[tid] - mean * a;
    }
    __syncthreads();
    float s = b_ih2[tid] + b_hh2[tid];
    for (int k = 0; k < kH; ++k) s += s_bb[k] * w_ih2[tid * kH + k];
    bbrow[tid] = s;
}

// -------------------------------------------------------------------------
// Layer-2 input projection (time-parallel WMMA GEMM):
//   xproj[row,:] = spk1[row,:] @ (diag(bn_a) w_ih2^T) + bbrow
// A tile = raw spike bytes -> async b128 copy of one contiguous 8 KB slab.
// -------------------------------------------------------------------------
__global__ __launch_bounds__(256) void proj2_gemm(
    const float* __restrict__ spk1,  // [T*B,128]
    const float* __restrict__ w_ih,  // [512,128]
    const float* __restrict__ bn_a,  // [128]
    const float* __restrict__ bbrow, // [512]
    float* __restrict__ xproj)       // [T*B,512] out
{
    __shared__ float lds_w[(kH / 2) * kPS];  // diag(a) * w_ih2^T, pair-interleaved
    __shared__ float lds_a[kBT * kAS];       // raw spike tile

    const int tid  = threadIdx.x;
    const int lane = tid & 31;
    const int wv   = tid >> 5;
    const int half = lane >> 4;
    const int l16  = lane & 15;

    for (int i = tid; i < kG * kH; i += 256) {
        int n = i / kH, k = i % kH;
        lds_w[(k >> 1) * kPS + 2 * n + (k & 1)] = bn_a[k] * w_ih[i];
    }
    float bias[4];
    for (int q = 0; q < 4; ++q) bias[q] = bbrow[(wv * 4 + q) * 16 + l16];
    __syncthreads();

    const int mtiles = kT * kB / kBT;  // 16000
    for (int mt = blockIdx.x; mt < mtiles; mt += gridDim.x) {
        // async-stage A tile: 16 rows x 128 floats, contiguous in spk1
        const float* gbase = spk1 + (size_t)mt * kBT * kH;
        #pragma unroll
        for (int c = tid; c < (kBT * kH) / 4; c += 256) {   // 512 x b128 chunks
            int r = c >> 5, k4 = (c & 31) * 4;
            async_b128(lds_off(&lds_a[r * kAS + k4]), gbase + c * 4);
        }
        async_wait0();
        __syncthreads();

        v8f acc[4];
        for (int q = 0; q < 4; ++q)
            for (int r = 0; r < 8; ++r) acc[q][r] = bias[q];

        #pragma unroll 4
        for (int ks = 0; ks < 32; ++ks) {
            v2f a = *(const v2f*)&lds_a[l16 * kAS + ks * 4 + 2 * half];
            #pragma unroll
            for (int q = 0; q < 4; ++q) {
                int n = (wv * 4 + q) * 16 + l16;
                v2f b = *(const v2f*)&lds_w[(ks * 2 + half) * kPS + 2 * n];
                acc[q] = wmma4(a, b, acc[q]);
            }
        }
        #pragma unroll
        for (int q = 0; q < 4; ++q)
            for (int r = 0; r < 8; ++r)
                xproj[(size_t)(mt * kBT + r + 8 * half) * kG + (wv * 4 + q) * 16 + l16] = acc[q][r];
        __syncthreads();  // protect lds_a before next tile
    }
}

// -------------------------------------------------------------------------
// Layer-2 SLSTM recurrence (records membrane potential mean over T).
// -------------------------------------------------------------------------
__global__ __launch_bounds__(256) void slstm_layer2(
    const float* __restrict__ xproj,  // [T*B,512] (bias folded in)
    const float* __restrict__ w_hh,   // [512,128]
    const float* __restrict__ thr_p,  // [1]
    float* __restrict__ mem_mean)     // [B,128] out
{
    __shared__ float lds_whh[(kH / 2) * kPS];
    __shared__ float lds_mem[kBT * kAS];

    const int tid  = threadIdx.x;
    const int lane = tid & 31;
    const int wv   = tid >> 5;
    const int half = lane >> 4;
    const int l16  = lane & 15;
    const int b0   = blockIdx.x * kBT;
    const float thr = thr_p[0];

    for (int i = tid; i < kG * kH; i += 256) {
        int n = i / kH, k = i % kH;
        lds_whh[(k >> 1) * kPS + 2 * n + (k & 1)] = w_hh[i];
    }
    for (int i = tid; i < kBT * kAS; i += 256) lds_mem[i] = 0.0f;

    float syn[8], memv[8], macc[8];
    for (int r = 0; r < 8; ++r) { syn[r] = 0.0f; memv[r] = 0.0f; macc[r] = 0.0f; }
    __syncthreads();

    for (int t = 0; t < kT; ++t) {
        v8f acc[4];
        #pragma unroll
        for (int q = 0; q < 4; ++q) {
            int col = q * kH + wv * 16 + l16;
            #pragma unroll
            for (int r = 0; r < 8; ++r)
                acc[q][r] = xproj[(size_t)(t * kB + b0 + r + 8 * half) * kG + col];
        }
        if (t + 1 < kT)  // prefetch next step's gate slab -> global_prefetch_b8
            __builtin_prefetch(&xproj[(size_t)((t + 1) * kB + b0 + 8 * half) * kG + wv * 16 + l16], 0, 0);

        #pragma unroll 4
        for (int ks = 0; ks < 32; ++ks) {
            v2f a = *(const v2f*)&lds_mem[l16 * kAS + ks * 4 + 2 * half];
            #pragma unroll
            for (int q = 0; q < 4; ++q) {
                int n = q * kH + wv * 16 + l16;
                v2f b = *(const v2f*)&lds_whh[(ks * 2 + half) * kPS + 2 * n];
                acc[q] = wmma4(a, b, acc[q]);
            }
        }
        __syncthreads();

        #pragma unroll
        for (int r = 0; r < 8; ++r) {
            float ig = sigmf(acc[0][r]);
            float fg = sigmf(acc[1][r]);
            float gg = tanhf(acc[2][r]);
            float og = sigmf(acc[3][r]);
            float s  = fg * syn[r] + ig * gg;
            float reset = (memv[r] - thr > 0.0f) ? thr : 0.0f;
            float mnew  = og * tanhf(s) - reset;
            syn[r] = s; memv[r] = mnew;
            macc[r] += mnew;
            lds_mem[(r + 8 * half) * kAS + wv * 16 + l16] = mnew;
        }
        __syncthreads();
    }
    for (int r = 0; r < 8; ++r)
        mem_mean[(size_t)(b0 + r + 8 * half) * kH + wv * 16 + l16] = macc[r] * (1.0f / (float)kT);
}

// -------------------------------------------------------------------------
__global__ void zero_kernel(float* __restrict__ p, int n) {
    int i = blockIdx.x * blockDim.x + threadIdx.x;
    if (i < n) p[i] = 0.0f;
}

__global__ __launch_bounds__(256) void fc_kernel(
    const float* __restrict__ mem_mean, const float* __restrict__ fc_w,
    const float* __restrict__ fc_b, float* __restrict__ out)
{
    int b = threadIdx.x;  // 256 threads, one per batch row
    float acc[7];
    for (int c = 0; c < 7; ++c) acc[c] = fc_b[c];
    for (int k = 0; k < kH; ++k) {
        float m = mem_mean[(size_t)b * kH + k];
        #pragma unroll
        for (int c = 0; c < 7; ++c) acc[c] += m * fc_w[c * kH + k];
    }
    for (int c = 0; c < 7; ++c) out[b * 7 + c] = acc[c];
}

// -------------------------------------------------------------------------
extern "C" void kernel_launch(void* const* d_in, const int* in_sizes, int n_in,
                              void* d_out, int out_size, void* d_ws, size_t ws_size,
                              hipStream_t stream) {
    (void)in_sizes; (void)n_in; (void)out_size; (void)ws_size;
    const float* x      = (const float*)d_in[0];
    const float* w_ih1  = (const float*)d_in[1];
    const float* w_hh1  = (const float*)d_in[2];
    const float* b_ih1  = (const float*)d_in[3];
    const float* b_hh1  = (const float*)d_in[4];
    const float* thr1   = (const float*)d_in[5];
    const float* w_ih2  = (const float*)d_in[6];
    const float* w_hh2  = (const float*)d_in[7];
    const float* b_ih2  = (const float*)d_in[8];
    const float* b_hh2  = (const float*)d_in[9];
    const float* thr2   = (const float*)d_in[10];
    const float* gamma  = (const float*)d_in[11];
    const float* beta   = (const float*)d_in[12];
    const float* fc_w   = (const float*)d_in[13];
    const float* fc_b   = (const float*)d_in[14];

    float* ws = (float*)d_ws;
    float* spk1   = ws;                                    // T*B*H
    float* xproj  = spk1  + (size_t)kT * kB * kH;          // T*B*4H
    float* bn_sum = xproj + (size_t)kT * kB * kG;          // H
    float* bn_a   = bn_sum + kH;                           // H
    float* bbrow  = bn_a + kH;                             // 4H
    float* memm   = bbrow + kG;                            // B*H

    zero_kernel<<<1, 128, 0, stream>>>(bn_sum, kH);
    slstm_layer1<<<kNWG, 256, 0, stream>>>(x, w_ih1, w_hh1, b_ih1, b_hh1, thr1, spk1, bn_sum);
    bn_proj_bias<<<1, 512, 0, stream>>>(bn_sum, gamma, beta, b_ih2, b_hh2, w_ih2, bn_a, bbrow);
    proj2_gemm<<<512, 256, 0, stream>>>(spk1, w_ih2, bn_a, bbrow, xproj);
    slstm_layer2<<<kNWG, 256, 0, stream>>>(xproj, w_hh2, thr2, memm);
    fc_kernel<<<1, 256, 0, stream>>>(memm, fc_w, fc_b, (float*)d_out);
}